// AdaptiveKernelFC_19542101196949
// MI455X (gfx1250) — compile-verified
//
#include <hip/hip_runtime.h>

// ---------------------------------------------------------------------------
// AdaptiveKernelFC on MI455X (gfx1250, wave32)
//
//   y[n,p,i,j] = Wk[p]*S1[n,i,j] + bk[p]*S2[n,i,j] + b_adap[p]
//   S1 = channel-summed spatial autocorrelation of x[n]  (from Gram G = X^T X)
//   S2 = rectangle sums of T[s] = sum_c x[n,c,s]
//
// G (49x49, K=256) via V_WMMA_F32_16X16X4_F32 on 4x4 tiles of 16x16.
// Padding rows/cols of G are garbage-by-design (clamped addresses): they only
// reach D elements the scatter discards, so the K-loop is branch-free and
// EXEC stays all-ones around every WMMA.
//
// LDS X^T layout [spatial][channel], row stride 260 words:
//   260 % 64 == 4  =>  fragment ds_load_b64: half-0 lanes use bank pairs
//   {4i,4i+1}, half-1 lanes {4i+2,4i+3} -> conflict-free across all 32 lanes.
// ---------------------------------------------------------------------------

typedef float v2f __attribute__((ext_vector_type(2)));
typedef float v8f __attribute__((ext_vector_type(8)));

#define CC 256      // channels = WMMA K extent
#define WW 7
#define SS 49       // spatial size
#define PP 256      // output planes
#define XS 260      // LDS row stride (words) for Xt[s][c]

__global__ __launch_bounds__(256)
void adaptive_fc_gram_wmma(const float* __restrict__ x,
                           const float* __restrict__ Wk,
                           const float* __restrict__ bk,
                           const float* __restrict__ badap,
                           float* __restrict__ out) {
    __shared__ __align__(16) float Xt[SS * XS];   // ~50 KB: X^T [s][c]
    __shared__ float Tl[64];
    __shared__ float S1[64];
    __shared__ float S2[64];

    const int tid  = threadIdx.x;
    const int n    = blockIdx.x;
    const int lane = tid & 31;
    const int wave = tid >> 5;

    __builtin_prefetch(Wk + tid, 0, 0);     // global_prefetch_b8
    __builtin_prefetch(bk + tid, 0, 0);
    __builtin_prefetch(badap + tid, 0, 0);

    // ---- Phase 1: zero S1, stage x[n] transposed into LDS ------------------
    if (tid < 64) S1[tid] = 0.0f;
    {
        const float* xg = x + (size_t)n * (CC * SS);
        int idx = tid;
        int c = tid / SS;              // 0..5
        int s = tid - c * SS;
        #pragma unroll 7
        for (int it = 0; it < (CC * SS) / 256; ++it) {   // 49 iterations
            Xt[s * XS + c] = xg[idx];
            idx += 256; c += 5; s += 11;                 // 256 = 5*49 + 11
            if (s >= SS) { s -= SS; ++c; }
        }
    }
    __syncthreads();

    // ---- Phase 2a: T[s] = sum_c Xt[s][c] -----------------------------------
    if (tid < SS) {
        float acc = 0.0f;
        #pragma unroll 8
        for (int c = 0; c < CC; ++c) acc += Xt[tid * XS + c];
        Tl[tid] = acc;
    }

    // ---- Phase 2b: Gram tiles via WMMA, branch-free K-loop -----------------
    // A frag (16x4 f32): lane%16 = M, K = (lane/16)*2 + vgpr  [ISA 7.12.2]
    // D frag: N = lane%16, M = vgpr + 8*(lane/16). G symmetric -> layout-safe.
    const int half = lane >> 4;
    const int sub  = lane & 15;
    const int mt   = wave >> 1;            // wave's row-tile
    const int nt0  = (wave & 1) * 2;       // two col-tiles nt0, nt0+1
    const int m    = mt * 16 + sub;
    const int n0   = nt0 * 16 + sub;
    const int n1   = n0 + 16;
    const int mc   = (m  < SS) ? m  : SS - 1;   // clamp: garbage rows discarded
    const int n0c  = (n0 < SS) ? n0 : SS - 1;
    const int n1c  = (n1 < SS) ? n1 : SS - 1;
    const float* pa  = &Xt[mc  * XS + half * 2];
    const float* pb0 = &Xt[n0c * XS + half * 2];
    const float* pb1 = &Xt[n1c * XS + half * 2];

    v8f acc0 = {}, acc1 = {};
    #pragma unroll
    for (int k = 0; k < CC; k += 4) {      // 64 steps: 3x ds_load_b64 + 2x wmma
        v2f a  = *reinterpret_cast<const v2f*>(pa  + k);
        v2f b0 = *reinterpret_cast<const v2f*>(pb0 + k);
        v2f b1 = *reinterpret_cast<const v2f*>(pb1 + k);
        acc0 = __builtin_amdgcn_wmma_f32_16x16x4_f32(
                   false, a, false, b0, (short)0, acc0, false, false);
        acc1 = __builtin_amdgcn_wmma_f32_16x16x4_f32(
                   false, a, false, b1, (short)0, acc1, false, false);
    }

    // Scatter G[s1][s2] into S1 at lag (s2-s1) when |di|,|dj| <= 3.
    #pragma unroll
    for (int r = 0; r < 8; ++r) {
        const int s1 = mt * 16 + r + 8 * half;
        if (s1 < SS) {
            const int u = s1 / WW, v = s1 - u * WW;
            if (n0 < SS) {
                int di = n0 / WW - u, dj = n0 % WW - v;
                if (di >= -3 && di <= 3 && dj >= -3 && dj <= 3)
                    atomicAdd(&S1[(di + 3) * WW + (dj + 3)], acc0[r]);
            }
            if (n1 < SS) {
                int di = n1 / WW - u, dj = n1 % WW - v;
                if (di >= -3 && di <= 3 && dj >= -3 && dj <= 3)
                    atomicAdd(&S1[(di + 3) * WW + (dj + 3)], acc1[r]);
            }
        }
    }
    __syncthreads();

    // ---- Phase 3: S2 rectangles from T -------------------------------------
    if (tid < SS) {
        int di = tid / WW - 3, dj = tid % WW - 3;
        int u0 = di > 0 ? di : 0, u1 = di < 0 ? 6 + di : 6;
        int v0 = dj > 0 ? dj : 0, v1 = dj < 0 ? 6 + dj : 6;
        float acc = 0.0f;
        for (int u = u0; u <= u1; ++u)
            for (int v = v0; v <= v1; ++v)
                acc += Tl[u * WW + v];
        S2[tid] = acc;
    }
    __syncthreads();

    // ---- Phase 4: fused coalesced output write -----------------------------
    {
        float* og = out + (size_t)n * (PP * SS);
        int idx = tid;
        int p = tid / SS;
        int s = tid - p * SS;
        #pragma unroll 7
        for (int it = 0; it < (PP * SS) / 256; ++it) {   // 49 iterations
            og[idx] = Wk[p] * S1[s] + bk[p] * S2[s] + badap[p];
            idx += 256; p += 5; s += 11;
            if (s >= SS) { s -= SS; ++p; }
        }
    }
}

extern "C" void kernel_launch(void* const* d_in, const int* in_sizes, int n_in,
                              void* d_out, int out_size, void* d_ws, size_t ws_size,
                              hipStream_t stream) {
    (void)n_in; (void)out_size; (void)d_ws; (void)ws_size;
    const float* x     = (const float*)d_in[0];
    const float* Wk    = (const float*)d_in[1];
    const float* bk    = (const float*)d_in[2];
    const float* badap = (const float*)d_in[3];
    float* out = (float*)d_out;

    const int N = in_sizes[0] / (CC * SS);   // 32 samples -> 32 workgroups
    adaptive_fc_gram_wmma<<<dim3(N), dim3(256), 0, stream>>>(x, Wk, bk, badap, out);
}